// WaveletLoss_26714696581389
// MI455X (gfx1250) — compile-verified
//
#include <hip/hip_runtime.h>

// ---------------------------------------------------------------------------
// WaveletLoss on MI455X (gfx1250): HBM-bound streaming kernel.
//   loss = mean|d| + sum_{l=1..3} ( mean|H_l(d)| + mean|V_l(d)| + mean|D_l(d)| )
// with d = pred - target (DWT is linear). Each lane owns one 8x8 tile; all
// 3 Haar levels are computed in-register. Reduction: WMMA ones-trick fold
// (32->16 lanes) + shfl_xor + LDS + deterministic second kernel.
// ---------------------------------------------------------------------------

typedef float v2f __attribute__((ext_vector_type(2)));
typedef float v8f __attribute__((ext_vector_type(8)));

// Full 32-lane sum, result broadcast to every lane.
// Stage 1: V_WMMA_F32_16X16X4_F32 with A=ones(16x4), B vgpr0 = x, vgpr1 = 0:
//   D[m][n] = sum_k B[k][n] = x[n] + x[n+16]  (column sum; layout-robust).
//   Every lane's D vgpr0 holds x[l%16] + x[l%16+16].
// Stage 2: xor-shuffle over masks 1,2,4,8 sums the 16 folded values.
// Requires EXEC all-ones (callers guarantee no divergence).
__device__ __forceinline__ float wave_reduce_full(float x) {
  v2f a; a[0] = 1.0f; a[1] = 1.0f;        // A = ones
  v2f b; b[0] = x;    b[1] = 0.0f;        // B rows: data + zeros
  v8f c = {};
  c = __builtin_amdgcn_wmma_f32_16x16x4_f32(
      /*neg_a=*/false, a, /*neg_b=*/false, b,
      /*c_mod=*/(short)0, c, /*reuse_a=*/false, /*reuse_b=*/false);
  float v = c[0];                          // = x[l%16] + x[l%16+16]
  v += __shfl_xor(v, 1, 32);
  v += __shfl_xor(v, 2, 32);
  v += __shfl_xor(v, 4, 32);
  v += __shfl_xor(v, 8, 32);
  return v;
}

// Shape: (32,3,512,512) -> 96 channels of 512x512. 8x8 tiles: 64x64 per
// channel, 393216 tiles total = 1536 blocks * 256 threads exactly.
#define IMG_W    512
#define TILES_X  64
#define N0 25165824.0f   // 32*3*512*512
#define N1 6291456.0f    // level-1 subband size
#define N2 1572864.0f
#define N3 393216.0f

__global__ __launch_bounds__(256) void wavelet_partial_kernel(
    const float* __restrict__ pred, const float* __restrict__ targ,
    float* __restrict__ partials) {
  const int tid    = blockIdx.x * 256 + threadIdx.x;
  const int tile_x = tid & 63;
  const int tile_y = (tid >> 6) & 63;
  const int chan   = tid >> 12;                 // 0..95
  const size_t base = (size_t)chan * (IMG_W * IMG_W)
                    + (size_t)(tile_y * 8) * IMG_W
                    + (size_t)(tile_x * 8);

  float s0 = 0.0f, s1 = 0.0f, s2 = 0.0f, s3 = 0.0f;
  float A1[2][4];   // two rows of level-1 cA (streamed)
  float A2[2][2];   // two rows of level-2 cA

  #pragma unroll
  for (int rp = 0; rp < 4; ++rp) {              // 4 row-pairs of the 8x8 tile
    float d[2][8];
    #pragma unroll
    for (int rr = 0; rr < 2; ++rr) {
      const size_t rb = base + (size_t)(rp * 2 + rr) * IMG_W;
      const float4 pa = *reinterpret_cast<const float4*>(pred + rb);
      const float4 pb = *reinterpret_cast<const float4*>(pred + rb + 4);
      const float4 ta = *reinterpret_cast<const float4*>(targ + rb);
      const float4 tb = *reinterpret_cast<const float4*>(targ + rb + 4);
      d[rr][0] = pa.x - ta.x; d[rr][1] = pa.y - ta.y;
      d[rr][2] = pa.z - ta.z; d[rr][3] = pa.w - ta.w;
      d[rr][4] = pb.x - tb.x; d[rr][5] = pb.y - tb.y;
      d[rr][6] = pb.z - tb.z; d[rr][7] = pb.w - tb.w;
      #pragma unroll
      for (int j = 0; j < 8; ++j) s0 += fabsf(d[rr][j]);
    }
    // Level 1: four 2x2 quads -> one row of cA + |H|+|V|+|D|
    #pragma unroll
    for (int j = 0; j < 4; ++j) {
      const float a = d[0][2*j], b = d[0][2*j+1];
      const float c = d[1][2*j], e = d[1][2*j+1];
      const float apb = a + b, amb = a - b, cpe = c + e, cme = c - e;
      A1[rp & 1][j] = 0.5f * (apb + cpe);
      s1 += 0.5f * (fabsf(apb - cpe) + fabsf(amb + cme) + fabsf(amb - cme));
    }
    // Level 2: every second row-pair completes two level-1 cA rows
    if (rp & 1) {
      const int q = rp >> 1;
      #pragma unroll
      for (int j = 0; j < 2; ++j) {
        const float a = A1[0][2*j], b = A1[0][2*j+1];
        const float c = A1[1][2*j], e = A1[1][2*j+1];
        const float apb = a + b, amb = a - b, cpe = c + e, cme = c - e;
        A2[q][j] = 0.5f * (apb + cpe);
        s2 += 0.5f * (fabsf(apb - cpe) + fabsf(amb + cme) + fabsf(amb - cme));
      }
    }
  }
  // Level 3: single 2x2 quad of level-2 cA
  {
    const float a = A2[0][0], b = A2[0][1], c = A2[1][0], e = A2[1][1];
    const float apb = a + b, amb = a - b, cpe = c + e, cme = c - e;
    s3 = 0.5f * (fabsf(apb - cpe) + fabsf(amb + cme) + fabsf(amb - cme));
  }

  const float part = s0 * (1.0f / N0) + s1 * (1.0f / N1)
                   + s2 * (1.0f / N2) + s3 * (1.0f / N3);

  // Wave reduce (WMMA fold + shuffles), then LDS across the 8 waves.
  const float wsum = wave_reduce_full(part);
  __shared__ float lds[8];
  const int lane = threadIdx.x & 31;
  const int wave = threadIdx.x >> 5;
  if (lane == 0) lds[wave] = wsum;
  __syncthreads();
  if (threadIdx.x == 0) {
    float t = 0.0f;
    #pragma unroll
    for (int w = 0; w < 8; ++w) t += lds[w];
    partials[blockIdx.x] = t;
  }
}

// Deterministic final reduction of 1536 block partials (one block; every
// thread does exactly 6 strided adds -> no divergence, fixed order).
__global__ __launch_bounds__(256) void wavelet_final_kernel(
    const float* __restrict__ partials, float* __restrict__ out, int n) {
  float acc = 0.0f;
  for (int i = threadIdx.x; i < n; i += 256) acc += partials[i];
  const float wsum = wave_reduce_full(acc);
  __shared__ float lds[8];
  const int lane = threadIdx.x & 31;
  const int wave = threadIdx.x >> 5;
  if (lane == 0) lds[wave] = wsum;
  __syncthreads();
  if (threadIdx.x == 0) {
    float t = 0.0f;
    #pragma unroll
    for (int w = 0; w < 8; ++w) t += lds[w];
    out[0] = t;
  }
}

extern "C" void kernel_launch(void* const* d_in, const int* in_sizes, int n_in,
                              void* d_out, int out_size, void* d_ws, size_t ws_size,
                              hipStream_t stream) {
  const float* pred = (const float*)d_in[0];
  const float* targ = (const float*)d_in[1];
  float* out       = (float*)d_out;
  float* partials  = (float*)d_ws;      // 1536 floats of scratch

  const int nblocks = 1536;             // 393216 tiles / 256 lanes per block
  wavelet_partial_kernel<<<nblocks, 256, 0, stream>>>(pred, targ, partials);
  wavelet_final_kernel<<<1, 256, 0, stream>>>(partials, out, nblocks);
}